// FaceEncoderEmbedding_68453188763948
// MI455X (gfx1250) — compile-verified
//
#include <hip/hip_runtime.h>

// FaceEncoderEmbedding: gather 3 embedding tables, add, scale by sqrt(256)=16,
// sum groups of 3 tokens -> out (B=16, K=2048, D=256) fp32.
//
// Memory-bound kernel (~0.03 flop/byte). MI455X strategy:
//  - coord table (4 KB) staged into LDS via CDNA5 async global->LDS copy
//    (global_load_async_to_lds_b128, ASYNCcnt, s_wait_asynccnt).
//  - token indices are wave-uniform -> readfirstlane so they load as scalar
//    (constant cache, KMcnt) and overlap with VMEM gathers.
//  - value/pos gathers stay in VMEM (tables are L2-resident, 192 MB L2).
//  - output written with non-temporal b128 stores (written once, never reread).

typedef float v4f __attribute__((ext_vector_type(4)));
typedef int   v4i __attribute__((ext_vector_type(4)));

#define BB   16
#define LLEN 6145
#define KK   2048
#define DD   256
#define ROWS_PER_BLOCK 4   // 256 threads / (DD/4 lanes per row)

__global__ __launch_bounds__(256) void face_embed_sum3_kernel(
    const int* __restrict__ val_tok,     // (B, L) int32
    const int* __restrict__ coord_tok,   // (B, L) int32
    const int* __restrict__ pos_tok,     // (B, L) int32
    const float* __restrict__ val_tab,   // (259, 256) f32
    const float* __restrict__ coord_tab, // (4, 256)   f32
    const float* __restrict__ pos_tab,   // (1000, 256) f32
    float* __restrict__ out)             // (B, K, 256) f32
{
    __shared__ float lds_coord[4 * DD];  // 4 KB: full coord_type_table

    const int tid = threadIdx.x;

    // ---- Stage coord table (1024 floats) into LDS: one async b128/thread ----
#if __has_builtin(__builtin_amdgcn_global_load_async_to_lds_b128)
    __builtin_amdgcn_global_load_async_to_lds_b128(
        (v4i*)(coord_tab + tid * 4),          // global source
        (v4i*)(&lds_coord[tid * 4]),          // LDS destination
        0, 0);
  #if __has_builtin(__builtin_amdgcn_s_wait_asynccnt)
    __builtin_amdgcn_s_wait_asynccnt(0);
  #else
    asm volatile("s_wait_asynccnt 0" ::: "memory");
  #endif
#else
    ((v4f*)lds_coord)[tid] = ((const v4f*)coord_tab)[tid];
#endif
    __syncthreads();

    // ---- Each group of 64 lanes (2 waves) handles one output row (b,k) ----
    // row is wave-uniform: force it into an SGPR so index loads scalarize.
    const int row  = __builtin_amdgcn_readfirstlane(
                         blockIdx.x * ROWS_PER_BLOCK + (tid >> 6));  // 0..32767
    const int lane = tid & 63;
    const int d0   = lane * 4;

    const int b = row >> 11;        // row / KK
    const int k = row & (KK - 1);   // row % KK
    const int tok_base = b * LLEN + 3 * k;   // scalar

    v4f acc = {0.0f, 0.0f, 0.0f, 0.0f};

#pragma unroll
    for (int j = 0; j < 3; ++j) {
        // Scalar loads through the constant cache (KMcnt), one per wave.
        const int vi = val_tok[tok_base + j];
        const int ci = coord_tok[tok_base + j];
        const int pi = pos_tok[tok_base + j];

        const v4f v = *(const v4f*)(val_tab + (size_t)vi * DD + d0);
        const v4f p = *(const v4f*)(pos_tab + (size_t)pi * DD + d0);
        const v4f c = *(const v4f*)(&lds_coord[ci * DD + d0]);

        acc.x += v.x + p.x + c.x;
        acc.y += v.y + p.y + c.y;
        acc.z += v.z + p.z + c.z;
        acc.w += v.w + p.w + c.w;
    }

    const float scale = 16.0f;  // sqrt(256)
    v4f res;
    res.x = acc.x * scale;
    res.y = acc.y * scale;
    res.z = acc.z * scale;
    res.w = acc.w * scale;

    // Non-temporal store: 32 MB output streamed once, keep it out of caches.
    v4f* dst = (v4f*)(out + (size_t)row * DD + d0);
    __builtin_nontemporal_store(res, dst);
}

extern "C" void kernel_launch(void* const* d_in, const int* in_sizes, int n_in,
                              void* d_out, int out_size, void* d_ws, size_t ws_size,
                              hipStream_t stream) {
    (void)in_sizes; (void)n_in; (void)d_ws; (void)ws_size; (void)out_size;

    const int*   val_tok   = (const int*)d_in[0];
    const int*   coord_tok = (const int*)d_in[1];
    const int*   pos_tok   = (const int*)d_in[2];
    const float* val_tab   = (const float*)d_in[3];
    const float* coord_tab = (const float*)d_in[4];
    const float* pos_tab   = (const float*)d_in[5];
    float*       out       = (float*)d_out;

    const int total_rows = BB * KK;                 // 32768
    const int blocks = total_rows / ROWS_PER_BLOCK; // 8192

    face_embed_sum3_kernel<<<blocks, 256, 0, stream>>>(
        val_tok, coord_tok, pos_tok, val_tab, coord_tab, pos_tab, out);
}